// MLPPredictor_60395830116585
// MI455X (gfx1250) — compile-verified
//
#include <hip/hip_runtime.h>

// MI455X / gfx1250 — hypernetwork MLP.
// Memory-bound: 673 MB of single-use per-problem weights streamed once
// (roofline floor ~29 us @ 23.3 TB/s; 337 MFLOP is negligible).
// FP32 WMMA (V_WMMA_F32_16X16X4_F32): weights flow through the B operand in
// native WMMA layout (64B-coalesced half-wave runs, each byte fetched exactly
// once -> non-temporal hint), activations broadcast into all 16 M-rows via
// ds_load_b64 so D row 0 is the GEMV result.
// Two wave32s cooperate per (s,b) problem (tile-split) to double resident
// waves (~8/SIMD) and keep >=14 MB of loads in flight for full HBM bandwidth.

typedef __attribute__((ext_vector_type(2))) float v2f;
typedef __attribute__((ext_vector_type(8))) float v8f;

#define S_DIM   128
#define B_DIM   32
#define X_DIM   32
#define H_DIM   128
#define Y_DIM   32
#define N_PROB  (S_DIM * B_DIM)                 // 4096 problems
#define W0_SIZE ((X_DIM + 1) * H_DIM)           // 4224
#define WI_SIZE (H_DIM * H_DIM)                 // 16384
#define Z_DIM   (W0_SIZE + 2 * WI_SIZE + H_DIM * Y_DIM) // 41088

#define PROBS_PER_BLOCK 4   // 8 waves/block, 2 waves per problem

// One layer slice: this wave produces output tiles [tbase, tbase+TILES).
// y_out[16*tile] = act( y_in[4*CHUNKS] @ W[4*CHUNKS x STRIDE-major] )
template <int CHUNKS, int TILES, int STRIDE, bool RELU>
__device__ __forceinline__ void mlp_layer(const float* __restrict__ w,
                                          const float* yin, float* yout,
                                          int tbase, int hi, int ln, int lane) {
  v2f a[CHUNKS];
#pragma unroll
  for (int c = 0; c < CHUNKS; ++c) {
    // lanes 0-15: (y[4c], y[4c+1]) = K0,K1 ; lanes 16-31: (y[4c+2], y[4c+3]) = K2,K3
    a[c] = *(const v2f*)(yin + 4 * c + 2 * hi);
  }
#pragma unroll
  for (int tt = 0; tt < TILES; ++tt) {
    const int t = tbase + tt;
    v8f acc = {};
    const float* wp = w + (2 * hi) * STRIDE + 16 * t + ln;
#pragma unroll
    for (int c = 0; c < CHUNKS; ++c) {
      v2f b;
      b.x = __builtin_nontemporal_load(wp);           // B row 4c+2hi,   col 16t+ln
      b.y = __builtin_nontemporal_load(wp + STRIDE);  // B row 4c+2hi+1, col 16t+ln
      wp += 4 * STRIDE;
      acc = __builtin_amdgcn_wmma_f32_16x16x4_f32(
          /*neg_a=*/false, a[c], /*neg_b=*/false, b,
          /*c_mod=*/(short)0, acc, /*reuse_a=*/false, /*reuse_b=*/false);
    }
    float r = acc[0];             // D row 0 (all rows identical by construction)
    if (RELU) r = fmaxf(r, 0.0f);
    if (lane < 16) yout[16 * t + ln] = r;
  }
}

__global__ __launch_bounds__(PROBS_PER_BLOCK * 64)
void hypernet_mlp_kernel(const float* __restrict__ x,
                         const float* __restrict__ z,
                         float* __restrict__ out) {
  __shared__ float ybuf[PROBS_PER_BLOCK][2][136]; // per-problem ping-pong activations
  const int lane = threadIdx.x & 31;
  const int wave = threadIdx.x >> 5;
  const int wsub = wave & 1;    // which half of the problem this wave owns
  const int pidx = wave >> 1;   // problem slot within the block
  const int hi   = lane >> 4;   // half-wave id: selects K pair (0,1) vs (2,3)
  const int ln   = lane & 15;   // N-column within the 16-wide tile
  const int prob = blockIdx.x * PROBS_PER_BLOCK + pidx;

  const float* zp = z + (size_t)prob * Z_DIM;
  float* yA = ybuf[pidx][0];
  float* yB = ybuf[pidx][1];

  // Stage x_ext = [x(32), 1, 0, 0, 0] -> K padded to 36 (9 chunks of 4).
  if (wsub == 0) {
    yA[lane] = __builtin_nontemporal_load(&x[(size_t)prob * X_DIM + lane]);
    if (lane < 4) yA[X_DIM + lane] = (lane == 0) ? 1.0f : 0.0f;
  }
  __syncthreads();

  // Layer 0: (33->36)x128. Rows 33..35 over-read finite z data but multiply by
  // the zero-padded activation entries -> contribute exactly 0.
  mlp_layer<9, 4, H_DIM, true>(zp, yA, yB, 4 * wsub, hi, ln, lane);
  __syncthreads();

  // Hidden layers: 128x128
  mlp_layer<32, 4, H_DIM, true>(zp + W0_SIZE, yB, yA, 4 * wsub, hi, ln, lane);
  __syncthreads();
  mlp_layer<32, 4, H_DIM, true>(zp + W0_SIZE + WI_SIZE, yA, yB, 4 * wsub, hi, ln, lane);
  __syncthreads();

  // Output layer: 128x32, no relu, write straight to global (1 tile per wave).
  mlp_layer<32, 1, Y_DIM, false>(zp + W0_SIZE + 2 * WI_SIZE, yB,
                                 out + (size_t)prob * Y_DIM, wsub, hi, ln, lane);
}

extern "C" void kernel_launch(void* const* d_in, const int* in_sizes, int n_in,
                              void* d_out, int out_size, void* d_ws, size_t ws_size,
                              hipStream_t stream) {
  (void)in_sizes; (void)n_in; (void)d_ws; (void)ws_size; (void)out_size;
  const float* x = (const float*)d_in[0];
  const float* z = (const float*)d_in[1];
  float* out = (float*)d_out;
  const int blocks = N_PROB / PROBS_PER_BLOCK; // 1024
  hipLaunchKernelGGL(hypernet_mlp_kernel, dim3(blocks), dim3(PROBS_PER_BLOCK * 64),
                     0, stream, x, z, out);
}